// LsunIntermediateRotation2dLayer_41601053229316
// MI455X (gfx1250) — compile-verified
//
#include <hip/hip_runtime.h>

typedef __attribute__((ext_vector_type(2))) float v2f;
typedef __attribute__((ext_vector_type(4))) float v4f;
typedef __attribute__((ext_vector_type(8))) float v8f;

#define PS 8
#define N_ANG 28
#define NBLKS (256 * 256)
#define NSAMP 64
#define NCH 16

// ---------------------------------------------------------------------------
// Kernel 1: build R per block from 28 cascaded Givens rotations + mu scaling,
// store it pre-swizzled into the WMMA B-operand (4x16 f32) lane layout:
//   slot (ks,v) holds VGPR v of the K-step ks B operand;
//   lane L: N = L%16 (output channel, zero for N>=8), K = 4*ks + v + 2*(L/16).
// ws layout: ws[b*128 + (ks*2+v)*32 + L]
// ---------------------------------------------------------------------------
__global__ __launch_bounds__(128) void build_R_kernel(
    const float* __restrict__ angles, const float* __restrict__ mus,
    float* __restrict__ ws) {
  const int b = blockIdx.x * blockDim.x + threadIdx.x;  // 0 .. NBLKS-1
  const float* a = angles + (size_t)b * N_ANG;

  float R[PS][PS];
#pragma unroll
  for (int i = 0; i < PS; ++i)
#pragma unroll
    for (int j = 0; j < PS; ++j) R[i][j] = (i == j) ? 1.0f : 0.0f;

  int k = 0;
#pragma unroll
  for (int t = 0; t < PS - 1; ++t) {
#pragma unroll
    for (int bb = t + 1; bb < PS; ++bb) {
      const float ang = a[k++];
      float s, c;
      __sincosf(ang, &s, &c);
#pragma unroll
      for (int i = 0; i < PS; ++i) {
        const float rt = R[t][i];
        const float rb = R[bb][i];
        R[t][i] = c * rt - s * rb;
        R[bb][i] = s * rt + c * rb;
      }
    }
  }

  // Row scaling by mu: R_final[j][i] = mu[j] * R[j][i]
#pragma unroll
  for (int j = 0; j < PS; ++j) {
    const float m = mus[(size_t)b * PS + j];
#pragma unroll
    for (int i = 0; i < PS; ++i) R[j][i] *= m;
  }

  float* w = ws + (size_t)b * 128;
#pragma unroll
  for (int ks = 0; ks < 2; ++ks) {
#pragma unroll
    for (int v = 0; v < 2; ++v) {
#pragma unroll
      for (int Lp = 0; Lp < 32; ++Lp) {
        const int col = Lp & 15;          // N (output channel)
        const int half = Lp >> 4;
        const int row = 4 * ks + v + 2 * half;  // K (input channel j)
        w[(ks * 2 + v) * 32 + Lp] = (col < PS) ? R[row][col] : 0.0f;
      }
    }
  }
}

// ---------------------------------------------------------------------------
// Kernel 2: streaming apply. One wave per block b; 4 waves per workgroup.
// D(16x16) = A(16x4) * B(4x16) + C, two K-steps covering K=8.
//   A: M = sample (lane%16 within a 16-sample tile), K = channel 8+4ks+koff+v
//   B: per-block rotation, pre-swizzled in ws
//   D: M = sample = v + (lane<16 ? 0 : 8), N = out channel 8 + lane%16 (N<8 valid)
// Channels 0..7 are a straight float4 pass-through copy.
// ---------------------------------------------------------------------------
__global__ __launch_bounds__(128) void apply_rot_kernel(
    const float* __restrict__ X, const float* __restrict__ ws,
    float* __restrict__ out) {
  const int lane = threadIdx.x & 31;
  const int wave = threadIdx.x >> 5;
  const int b = blockIdx.x * 4 + wave;  // spatial block index
  const int lmod = lane & 15;
  const int hi = lane >> 4;             // lane half
  const int koff = hi * 2;              // A-operand K offset within K-step

  // B operands: 4 fully coalesced 128B loads per wave
  const float* wsb = ws + (size_t)b * 128;
  v2f B0, B1;
  B0.x = wsb[0 * 32 + lane];
  B0.y = wsb[1 * 32 + lane];
  B1.x = wsb[2 * 32 + lane];
  B1.y = wsb[3 * 32 + lane];

  const size_t sampleStride = (size_t)NBLKS * NCH;  // floats per sample image
  const size_t pixBase = (size_t)b * NCH;

#pragma unroll
  for (int mt = 0; mt < 4; ++mt) {
    const int m = mt * 16 + lmod;  // sample index for loads
    const float* px = X + (size_t)m * sampleStride + pixBase;
    float* py = out + (size_t)m * sampleStride + pixBase;

    // Pass-through copy of channels 0..7 (lane halves cover ch 0-3 / 4-7)
    const int c0 = hi * 4;
    const v4f z = *(const v4f*)(px + c0);
    *(v4f*)(py + c0) = z;

    // A operands: ks=0 -> channels 8+koff..9+koff ; ks=1 -> 12+koff..13+koff
    const v2f A0 = *(const v2f*)(px + 8 + koff);
    const v2f A1 = *(const v2f*)(px + 12 + koff);

    v8f c = {};
    c = __builtin_amdgcn_wmma_f32_16x16x4_f32(false, A0, false, B0,
                                              (short)0, c, false, false);
    c = __builtin_amdgcn_wmma_f32_16x16x4_f32(false, A1, false, B1,
                                              (short)0, c, false, false);

    // Store valid D columns (N = lmod < 8): out channel 8+lmod of sample
    // mt*16 + v + (hi ? 8 : 0). Lanes 0-7 / 16-23 each store 8 dwords.
    if (lmod < PS) {
      const int mbase = mt * 16 + hi * 8;
#pragma unroll
      for (int v = 0; v < 8; ++v) {
        out[(size_t)(mbase + v) * sampleStride + pixBase + 8 + lmod] = c[v];
      }
    }
  }
}

extern "C" void kernel_launch(void* const* d_in, const int* in_sizes, int n_in,
                              void* d_out, int out_size, void* d_ws,
                              size_t ws_size, hipStream_t stream) {
  const float* X = (const float*)d_in[0];       // [64,256,256,16] f32
  const float* angles = (const float*)d_in[1];  // [65536,28] f32
  const float* mus = (const float*)d_in[2];     // [65536,8] f32
  float* out = (float*)d_out;                   // [64,256,256,16] f32
  float* ws = (float*)d_ws;                     // needs 65536*128*4 = 32 MB

  // Kernel 1: 65536 threads, one per spatial block
  build_R_kernel<<<NBLKS / 128, 128, 0, stream>>>(angles, mus, ws);

  // Kernel 2: one wave per spatial block, 4 waves (128 threads) per WG
  apply_rot_kernel<<<NBLKS / 4, 128, 0, stream>>>(X, ws, out);
}